// MF_MGCN_45741401702849
// MI455X (gfx1250) — compile-verified
//
#include <hip/hip_runtime.h>
#include <hip/hip_bf16.h>

// ---------------- problem constants ----------------
static constexpr int   kB   = 32768;
static constexpr int   kNN  = 19;
static constexpr int   kN   = kB * kNN;     // 622592 nodes  (divisible by 256)
static constexpr int   kEF  = kB * 120;     // 3,932,160 func edges
static constexpr int   kES  = kB * 60;      // 1,966,080 struct edges
static constexpr float kEPS = 1e-5f;

typedef float v2f __attribute__((ext_vector_type(2)));
typedef float v8f __attribute__((ext_vector_type(8)));

// ---------------- init: degrees to 1 (self loop), accumulators to 0 --------
__global__ void k_init(float* __restrict__ deg_f, float* __restrict__ deg_s,
                       float* __restrict__ aAgg, float* __restrict__ agg2,
                       float* __restrict__ stats) {
    size_t i = (size_t)blockIdx.x * blockDim.x + threadIdx.x;
    if (i < (size_t)kN)      { deg_f[i] = 1.0f; deg_s[i] = 1.0f; }
    if (i < (size_t)5 * kN)  aAgg[i] = 0.0f;
    if (i < (size_t)10 * kN) agg2[i] = 0.0f;
    if (i < 286)             stats[i] = 0.0f;   // stats1(10)+stats2(20)+stats3(256)
}

// ---------------- weighted in-degree for both edge sets -------------------
__global__ void k_deg(const int* __restrict__ dstF, const float* __restrict__ ewF,
                      const int* __restrict__ dstS,
                      float* __restrict__ deg_f, float* __restrict__ deg_s) {
    int e = blockIdx.x * blockDim.x + threadIdx.x;
    if (e < kEF) atomicAdd(&deg_f[dstF[e]], ewF[e]);
    if (e < kES) atomicAdd(&deg_s[dstS[e]], 1.0f);
}

__global__ void k_dinv(const float* __restrict__ deg_f, const float* __restrict__ deg_s,
                       float* __restrict__ dinv_f, float* __restrict__ dinv_s) {
    int n = blockIdx.x * blockDim.x + threadIdx.x;
    if (n < kN) {
        dinv_f[n] = rsqrtf(deg_f[n]);
        dinv_s[n] = rsqrtf(deg_s[n]);
    }
}

// ---------------- GCN1 collapsed to scalar scatter per band ----------------
__global__ void k_scatter1(const int* __restrict__ srcF, const int* __restrict__ dstF,
                           const float* __restrict__ ewF, const float* __restrict__ dinv_f,
                           const float* __restrict__ x, float* __restrict__ aAgg) {
    int e = blockIdx.x * blockDim.x + threadIdx.x;
    if (e >= kEF) return;
    int s = srcF[e], d = dstF[e];
    float w = dinv_f[s] * ewF[e] * dinv_f[d];
    const float* xs = x + (size_t)s * 5;
#pragma unroll
    for (int b = 0; b < 5; ++b)
        atomicAdd(&aAgg[(size_t)b * kN + d], w * xs[b]);
}

// self-loop term + per-band scalar BN1 statistics (wave32 shuffle reduction)
__global__ void k_self1(const float* __restrict__ x, const float* __restrict__ dinv_f,
                        float* __restrict__ aAgg, float* __restrict__ stats1) {
    int n = blockIdx.x * blockDim.x + threadIdx.x;
    bool ok = n < kN;
    float d2 = ok ? dinv_f[n] * dinv_f[n] : 0.0f;
#pragma unroll
    for (int b = 0; b < 5; ++b) {
        float v = 0.0f;
        if (ok) {
            size_t idx = (size_t)b * kN + n;
            v = aAgg[idx] + d2 * x[(size_t)n * 5 + b];
            aAgg[idx] = v;
        }
        float s = v, s2 = v * v;
#pragma unroll
        for (int m = 16; m > 0; m >>= 1) {
            s  += __shfl_xor(s,  m, 32);
            s2 += __shfl_xor(s2, m, 32);
        }
        if ((threadIdx.x & 31) == 0) {
            atomicAdd(&stats1[2 * b],     s);
            atomicAdd(&stats1[2 * b + 1], s2);
        }
    }
}

// tiny: alpha_c = g1*W1c*rsqrt(var(a)*W1c^2+eps)  (b1 cancels in BN)
__global__ void k_params1(const float* __restrict__ stats1, const float* __restrict__ W1,
                          const float* __restrict__ g1,
                          float* __restrict__ alpha, float* __restrict__ maArr) {
    int i = threadIdx.x;
    if (i < 160) {
        int b = i >> 5;
        float ma = stats1[2 * b] * (1.0f / kN);
        float va = stats1[2 * b + 1] * (1.0f / kN) - ma * ma;
        float w = W1[i];
        alpha[i] = g1[i] * w * rsqrtf(va * w * w + kEPS);
        if ((i & 31) == 0) maArr[b] = ma;
    }
}

// per-node GCN2 pre-aggregation: p_j(n) = sum_c relu(alpha*t+bt1) * W2[c,j]
__global__ __launch_bounds__(256)
void k_h2pre(const float* __restrict__ aAgg, const float* __restrict__ maArr,
             const float* __restrict__ alpha, const float* __restrict__ bt1,
             const float* __restrict__ W2, float* __restrict__ h2p) {
    __shared__ float sAl[160], sBt[160], sW2[320], sMa[5];
    for (int i = threadIdx.x; i < 160; i += 256) { sAl[i] = alpha[i]; sBt[i] = bt1[i]; }
    for (int i = threadIdx.x; i < 320; i += 256) sW2[i] = W2[i];
    if (threadIdx.x < 5) sMa[threadIdx.x] = maArr[threadIdx.x];
    __syncthreads();
    int n = blockIdx.x * 256 + threadIdx.x;
    if (n >= kN) return;
#pragma unroll
    for (int b = 0; b < 5; ++b) {
        float t = aAgg[(size_t)b * kN + n] - sMa[b];
        float p0 = 0.0f, p1 = 0.0f;
#pragma unroll
        for (int c = 0; c < 32; ++c) {
            float h = fmaxf(fmaf(sAl[b * 32 + c], t, sBt[b * 32 + c]), 0.0f);
            p0 = fmaf(h, sW2[(b * 32 + c) * 2],     p0);
            p1 = fmaf(h, sW2[(b * 32 + c) * 2 + 1], p1);
        }
        size_t o = ((size_t)b * kN + n) * 2;
        h2p[o] = p0; h2p[o + 1] = p1;
    }
}

// ---------------- GCN2 struct-edge scatter (2 channels x 5 bands) ----------
__global__ void k_scatter2(const int* __restrict__ srcS, const int* __restrict__ dstS,
                           const float* __restrict__ dinv_s,
                           const float* __restrict__ h2p, float* __restrict__ agg2) {
    int e = blockIdx.x * blockDim.x + threadIdx.x;
    if (e >= kES) return;
    int s = srcS[e], d = dstS[e];
    float w = dinv_s[s] * dinv_s[d];
#pragma unroll
    for (int b = 0; b < 5; ++b) {
        size_t so = ((size_t)b * kN + s) * 2;
        size_t dd = ((size_t)b * kN + d) * 2;
        atomicAdd(&agg2[dd],     w * h2p[so]);
        atomicAdd(&agg2[dd + 1], w * h2p[so + 1]);
    }
}

// self-loop + bias + BN2 statistics
__global__ void k_self2(const float* __restrict__ h2p, const float* __restrict__ dinv_s,
                        const float* __restrict__ b2, float* __restrict__ agg2,
                        float* __restrict__ stats2) {
    int n = blockIdx.x * blockDim.x + threadIdx.x;
    bool ok = n < kN;
    float d2 = ok ? dinv_s[n] * dinv_s[n] : 0.0f;
#pragma unroll
    for (int ch = 0; ch < 10; ++ch) {
        int b = ch >> 1, j = ch & 1;
        float v = 0.0f;
        if (ok) {
            size_t idx = ((size_t)b * kN + n) * 2 + j;
            v = agg2[idx] + d2 * h2p[idx] + b2[ch];
            agg2[idx] = v;
        }
        float s = v, s2 = v * v;
#pragma unroll
        for (int m = 16; m > 0; m >>= 1) {
            s  += __shfl_xor(s,  m, 32);
            s2 += __shfl_xor(s2, m, 32);
        }
        if ((threadIdx.x & 31) == 0) {
            atomicAdd(&stats2[2 * ch],     s);
            atomicAdd(&stats2[2 * ch + 1], s2);
        }
    }
}

__global__ void k_params2(const float* __restrict__ stats2, const float* __restrict__ g2,
                          const float* __restrict__ bt2,
                          float* __restrict__ bn2sc, float* __restrict__ bn2sh) {
    int i = threadIdx.x;
    if (i < 10) {
        float m = stats2[2 * i] * (1.0f / kN);
        float v = stats2[2 * i + 1] * (1.0f / kN) - m * m;
        float sc = g2[i] * rsqrtf(v + kEPS);
        bn2sc[i] = sc;
        bn2sh[i] = bt2[i] - m * sc;
    }
}

// build xc[B,192] (190 real cols, 2 zero pad), BN2+ReLU applied
__global__ void k_xc(const float* __restrict__ agg2, const float* __restrict__ bn2sc,
                     const float* __restrict__ bn2sh, float* __restrict__ xc) {
    int g = blockIdx.x, col = threadIdx.x;   // blockDim = 192
    float v = 0.0f;
    if (col < 190) {
        int b = col / 38, r = col - b * 38;
        int nn = r >> 1, j = r & 1;
        int ch = b * 2 + j;
        size_t idx = ((size_t)b * kN + (size_t)g * kNN + nn) * 2 + j;
        v = fmaxf(fmaf(agg2[idx], bn2sc[ch], bn2sh[ch]), 0.0f);
    }
    xc[(size_t)g * 192 + col] = v;
}

// ---------------- WMMA GEMM1: Y[B,128] = xc[B,192] @ lin1_W + b ------------
__global__ __launch_bounds__(256)
void k_gemm1(const float* __restrict__ xc, const float* __restrict__ W,
             const float* __restrict__ bias, float* __restrict__ Y) {
    __shared__ float Bs[192 * 16];                 // 12 KB B-tile
    const int colBase = blockIdx.y * 16;
    for (int i = threadIdx.x; i < 192 * 16; i += 256) {
        int k = i >> 4, n = i & 15;
        Bs[i] = (k < 190) ? W[k * 128 + colBase + n] : 0.0f;
    }
    __syncthreads();
    const int wave = threadIdx.x >> 5, lane = threadIdx.x & 31;
    const int lh = lane & 15, hi = lane >> 4;
    const int rowBase = blockIdx.x * 128 + wave * 16;
    const float* arow = xc + (size_t)(rowBase + lh) * 192;
    __builtin_prefetch(arow, 0, 1);
    v8f acc = {};
#pragma unroll 4
    for (int kk = 0; kk < 192; kk += 4) {
        int kA = kk + 2 * hi;
        v2f a = *(const v2f*)(arow + kA);          // 8B aligned (kA even)
        v2f b;
        b.x = Bs[kA * 16 + lh];
        b.y = Bs[(kA + 1) * 16 + lh];
        acc = __builtin_amdgcn_wmma_f32_16x16x4_f32(false, a, false, b,
                                                    (short)0, acc, false, false);
    }
    const int col = colBase + lh;
    const float bv = bias[col];
#pragma unroll
    for (int v = 0; v < 8; ++v) {
        int r = rowBase + hi * 8 + v;              // C layout: VGPR v -> M = v (+8 hi)
        Y[(size_t)r * 128 + col] = acc[v] + bv;
    }
}

// column mean/sumsq over Y for BN3
__global__ __launch_bounds__(256)
void k_colstats(const float* __restrict__ Y, float* __restrict__ stats3) {
    __shared__ float ss[256], ss2[256];
    int c = blockIdx.x;
    float s = 0.0f, s2 = 0.0f;
    for (int r = threadIdx.x; r < kB; r += 256) {
        float v = Y[(size_t)r * 128 + c];
        s += v; s2 = fmaf(v, v, s2);
    }
    ss[threadIdx.x] = s; ss2[threadIdx.x] = s2;
    __syncthreads();
    for (int step = 128; step > 0; step >>= 1) {
        if (threadIdx.x < step) {
            ss[threadIdx.x]  += ss[threadIdx.x + step];
            ss2[threadIdx.x] += ss2[threadIdx.x + step];
        }
        __syncthreads();
    }
    if (threadIdx.x == 0) { stats3[c] = ss[0]; stats3[128 + c] = ss2[0]; }
}

__global__ void k_params3(const float* __restrict__ stats3, const float* __restrict__ g3,
                          const float* __restrict__ bt3,
                          float* __restrict__ bn3sc, float* __restrict__ bn3sh) {
    int c = threadIdx.x;   // 128
    float m = stats3[c] * (1.0f / kB);
    float v = stats3[128 + c] * (1.0f / kB) - m * m;
    float sc = g3[c] * rsqrtf(v + kEPS);
    bn3sc[c] = sc;
    bn3sh[c] = bt3[c] - m * sc;
}

// --- WMMA GEMM2 fused: relu(bn3(Y)) @ lin2_W + b, relu, @ lin3_W + b → out -
__global__ __launch_bounds__(256)
void k_gemm2(const float* __restrict__ Y, const float* __restrict__ sc,
             const float* __restrict__ sh, const float* __restrict__ W2,
             const float* __restrict__ b2, const float* __restrict__ W3,
             const float* __restrict__ b3, float* __restrict__ out) {
    __shared__ float Zs[64 * 32];                  // 8 KB hidden tile
    const int wave = threadIdx.x >> 5, lane = threadIdx.x & 31;
    const int lh = lane & 15, hi = lane >> 4;
    const int rowTile = wave & 3, colTile = wave >> 2;
    const int rowBase = blockIdx.x * 64 + rowTile * 16;
    const int colBase = colTile * 16;
    const float* arow = Y + (size_t)(rowBase + lh) * 128;
    __builtin_prefetch(arow, 0, 1);
    v8f acc = {};
#pragma unroll 4
    for (int kk = 0; kk < 128; kk += 4) {
        int kA = kk + 2 * hi;
        v2f a;
        a.x = fmaxf(fmaf(arow[kA],     sc[kA],     sh[kA]),     0.0f);
        a.y = fmaxf(fmaf(arow[kA + 1], sc[kA + 1], sh[kA + 1]), 0.0f);
        v2f b;
        b.x = W2[kA * 32 + colBase + lh];
        b.y = W2[(kA + 1) * 32 + colBase + lh];
        acc = __builtin_amdgcn_wmma_f32_16x16x4_f32(false, a, false, b,
                                                    (short)0, acc, false, false);
    }
    const int col = colBase + lh;
    const float bv = b2[col];
#pragma unroll
    for (int v = 0; v < 8; ++v) {
        int rl = rowTile * 16 + hi * 8 + v;
        Zs[rl * 32 + col] = fmaxf(acc[v] + bv, 0.0f);
    }
    __syncthreads();
    if (threadIdx.x < 128) {
        int rl = threadIdx.x >> 1, j = threadIdx.x & 1;
        float s = b3[j];
#pragma unroll
        for (int c = 0; c < 32; ++c) s = fmaf(Zs[rl * 32 + c], W3[c * 2 + j], s);
        out[(size_t)(blockIdx.x * 64 + rl) * 2 + j] = s;
    }
}

// -------------------------------- launcher ---------------------------------
extern "C" void kernel_launch(void* const* d_in, const int* in_sizes, int n_in,
                              void* d_out, int out_size, void* d_ws, size_t ws_size,
                              hipStream_t stream) {
    // inputs in setup_inputs() dict order
    const float* x    = (const float*)d_in[0];         // [N,5]
    const float* ewF  = (const float*)d_in[1];         // [EF]
    const int*   eiF  = (const int*)d_in[2];           // [2,EF]
    const int*   eiS  = (const int*)d_in[3];           // [2,ES]
    const float* W1   = (const float*)d_in[4];         // [5,1,32]
    // b1 = d_in[5]  (cancels inside BN1 — unused)
    const float* g1   = (const float*)d_in[6];
    const float* bt1  = (const float*)d_in[7];
    const float* W2   = (const float*)d_in[8];         // [5,32,2]
    const float* b2   = (const float*)d_in[9];         // [5,2]
    const float* g2   = (const float*)d_in[10];
    const float* bt2  = (const float*)d_in[11];
    const float* l1W  = (const float*)d_in[12];        // [190,128]
    const float* l1b  = (const float*)d_in[13];
    const float* g3   = (const float*)d_in[14];
    const float* bt3  = (const float*)d_in[15];
    const float* l2W  = (const float*)d_in[16];        // [128,32]
    const float* l2b  = (const float*)d_in[17];
    const float* l3W  = (const float*)d_in[18];        // [32,2]
    const float* l3b  = (const float*)d_in[19];
    float* out = (float*)d_out;                        // [B,2]

    const int* srcF = eiF;        const int* dstF = eiF + kEF;
    const int* srcS = eiS;        const int* dstS = eiS + kES;

    // workspace carve-up
    float* ws     = (float*)d_ws;
    float* deg_f  = ws;                          // N
    float* deg_s  = deg_f  + kN;                 // N
    float* dinv_f = deg_s  + kN;                 // N
    float* dinv_s = dinv_f + kN;                 // N
    float* aAgg   = dinv_s + kN;                 // 5N
    float* h2p    = aAgg   + (size_t)5 * kN;     // 10N
    float* agg2   = h2p    + (size_t)10 * kN;    // 10N
    float* xc     = agg2   + (size_t)10 * kN;    // B*192
    float* Y      = xc     + (size_t)kB * 192;   // B*128
    float* stats  = Y      + (size_t)kB * 128;   // 286 (stats1|stats2|stats3)
    float* stats1 = stats;
    float* stats2 = stats + 10;
    float* stats3 = stats + 30;
    float* alpha  = stats + 286;                 // 160
    float* maArr  = alpha + 160;                 // 5
    float* bn2sc  = maArr + 5;                   // 10
    float* bn2sh  = bn2sc + 10;                  // 10
    float* bn3sc  = bn2sh + 10;                  // 128
    float* bn3sh  = bn3sc + 128;                 // 128

    const int T = 256;
    k_init     <<<dim3(((size_t)10 * kN + T - 1) / T), T, 0, stream>>>(deg_f, deg_s, aAgg, agg2, stats);
    k_deg      <<<dim3((kEF + T - 1) / T), T, 0, stream>>>(dstF, ewF, dstS, deg_f, deg_s);
    k_dinv     <<<dim3((kN + T - 1) / T), T, 0, stream>>>(deg_f, deg_s, dinv_f, dinv_s);
    k_scatter1 <<<dim3((kEF + T - 1) / T), T, 0, stream>>>(srcF, dstF, ewF, dinv_f, x, aAgg);
    k_self1    <<<dim3(kN / T), T, 0, stream>>>(x, dinv_f, aAgg, stats1);
    k_params1  <<<dim3(1), 160, 0, stream>>>(stats1, W1, g1, alpha, maArr);
    k_h2pre    <<<dim3(kN / T), T, 0, stream>>>(aAgg, maArr, alpha, bt1, W2, h2p);
    k_scatter2 <<<dim3((kES + T - 1) / T), T, 0, stream>>>(srcS, dstS, dinv_s, h2p, agg2);
    k_self2    <<<dim3(kN / T), T, 0, stream>>>(h2p, dinv_s, b2, agg2, stats2);
    k_params2  <<<dim3(1), 32, 0, stream>>>(stats2, g2, bt2, bn2sc, bn2sh);
    k_xc       <<<dim3(kB), 192, 0, stream>>>(agg2, bn2sc, bn2sh, xc);
    k_gemm1    <<<dim3(kB / 128, 8), T, 0, stream>>>(xc, l1W, l1b, Y);
    k_colstats <<<dim3(128), T, 0, stream>>>(Y, stats3);
    k_params3  <<<dim3(1), 128, 0, stream>>>(stats3, g3, bt3, bn3sc, bn3sh);
    k_gemm2    <<<dim3(kB / 64), T, 0, stream>>>(Y, bn3sc, bn3sh, l2W, l2b, l3W, l3b, out);
    (void)in_sizes; (void)n_in; (void)out_size; (void)ws_size;
}